// SPNNeuron_24309514895914
// MI455X (gfx1250) — compile-verified
//
#include <hip/hip_runtime.h>
#include <hip/hip_bf16.h>
#include <math.h>

// ---------------------------------------------------------------------------
// SPN neuron forward, MI455X (gfx1250, wave32).
//   out[b] = sum_p logsumexp_m( sum_leaf logN(x[b,scopes[2p+leaf]]; mu,sigma)
//                               + logw[p,m] )
// Per (p,m) the leaf-pair log-density is the polynomial
//   t = a0*x0^2 + b0*x0 + a1*x1^2 + b1*x1 + c   (coeffs pre-scaled by log2 e)
// evaluated with V_WMMA_F32_16X16X4_F32 (A = coeffs for 2 pairs x 8 comps,
// B = [x^2, x] features of 16 batch rows, C = constants), then a thread-local
// base-2 logsumexp over the 8 accumulator VGPRs (v_exp_f32 / v_log_f32).
// x tile staged to LDS with GLOBAL_LOAD_ASYNC_TO_LDS_B128.
// ---------------------------------------------------------------------------

#define B_TOTAL 32768
#define F_FEAT  512
#define P_PAIR  256
#define M_MIX   8
#define ROWS_PER_BLOCK 128
#define THREADS 256
#define XS_STRIDE 516   // 512 + 4 pad floats; 516 % 64 == 4 -> conflict-free column gather

#define L2E    1.4426950408889634f   // log2(e)
#define LN2    0.6931471805599453f   // ln(2)
#define LN_2PI 1.8378770664093453f   // ln(2*pi)

typedef __attribute__((ext_vector_type(2))) float v2f;
typedef __attribute__((ext_vector_type(8))) float v8f;
typedef int v4i __attribute__((vector_size(16)));

// LDS layout (bytes)
#define XS_OFF   0
#define XS_BYTES (ROWS_PER_BLOCK * XS_STRIDE * 4)   // 264192
#define CA_OFF   (XS_OFF + XS_BYTES)                // coefA[P][M][4] = {a0,b0,a1,b1}
#define CA_BYTES (P_PAIR * M_MIX * 4 * 4)           // 32768
#define CT_OFF   (CA_OFF + CA_BYTES)                // cTab[P][M]
#define CT_BYTES (P_PAIR * M_MIX * 4)               // 8192
#define SI_OFF   (CT_OFF + CT_BYTES)                // sidx[F]
#define SI_BYTES (F_FEAT * 4)                       // 2048
#define LDS_TOTAL (SI_OFF + SI_BYTES)               // 307200 B < 320KB WGP LDS

// CDNA5 async global->LDS path (guarded; fallback = load + ds_store)
#ifndef USE_ASYNC
#if defined(__has_builtin)
#if __has_builtin(__builtin_amdgcn_global_load_async_to_lds_b128) && \
    __has_builtin(__builtin_amdgcn_s_wait_asynccnt)
#define USE_ASYNC 1
#else
#define USE_ASYNC 0
#endif
#else
#define USE_ASYNC 0
#endif
#endif

// fast exp2/log2 that lower to v_exp_f32 / v_log_f32
#if defined(__has_builtin)
#if __has_builtin(__builtin_amdgcn_exp2f)
#define EXP2F(v) __builtin_amdgcn_exp2f(v)
#else
#define EXP2F(v) exp2f(v)
#endif
#if __has_builtin(__builtin_amdgcn_logf)
#define LOG2F(v) __builtin_amdgcn_logf(v)
#else
#define LOG2F(v) log2f(v)
#endif
#else
#define EXP2F(v) exp2f(v)
#define LOG2F(v) log2f(v)
#endif

__global__ __launch_bounds__(THREADS)
void spn_wmma_kernel(const float* __restrict__ x,
                     const float* __restrict__ mean,
                     const float* __restrict__ stdv,
                     const float* __restrict__ weights,
                     const long long* __restrict__ scopes,
                     float* __restrict__ out)
{
    extern __shared__ char smem[];
    float* xs    = (float*)(smem + XS_OFF);
    float* coefA = (float*)(smem + CA_OFF);
    float* cTab  = (float*)(smem + CT_OFF);
    int*   sidx  = (int*)  (smem + SI_OFF);

    const int tid = threadIdx.x;

    // ---- scope permutation (int64 -> int32) ----
    for (int f = tid; f < F_FEAT; f += THREADS)
        sidx[f] = (int)scopes[f];

    // ---- per-(p,m) polynomial coefficients, log2 domain ----
    for (int idx = tid; idx < P_PAIR * M_MIX; idx += THREADS) {
        const int p = idx >> 3;
        float mu0 = mean[idx * 2 + 0], mu1 = mean[idx * 2 + 1];
        float s0  = stdv[idx * 2 + 0], s1  = stdv[idx * 2 + 1];
        float wsum = 0.f;
#pragma unroll
        for (int j = 0; j < M_MIX; ++j) wsum += weights[p * M_MIX + j];
        float lw = __logf(weights[idx]) - __logf(wsum);
        float i0 = 1.f / (s0 * s0), i1 = 1.f / (s1 * s1);
        coefA[idx * 4 + 0] = -0.5f * i0 * L2E;   // K0 * x0^2
        coefA[idx * 4 + 1] =  mu0  * i0 * L2E;   // K1 * x0
        coefA[idx * 4 + 2] = -0.5f * i1 * L2E;   // K2 * x1^2
        coefA[idx * 4 + 3] =  mu1  * i1 * L2E;   // K3 * x1
        float cc = -0.5f * (mu0 * mu0 * i0 + mu1 * mu1 * i1)
                   - __logf(s0) - __logf(s1) - LN_2PI + lw;
        cTab[idx] = cc * L2E;
    }

    // ---- stage 128 batch rows of x into LDS (coalesced b128, async if available) ----
    const long long rowbase = (long long)blockIdx.x * ROWS_PER_BLOCK;
    {
        const float4* gx = (const float4*)(x + rowbase * F_FEAT);
        for (int i = tid; i < ROWS_PER_BLOCK * (F_FEAT / 4); i += THREADS) {
            const int r  = i >> 7;     // / (F/4 = 128)
            const int c4 = i & 127;
#if USE_ASYNC
            __builtin_amdgcn_global_load_async_to_lds_b128(
                (__attribute__((address_space(1))) v4i*)
                    (__attribute__((address_space(1))) void*)
                        (void*)(gx + (size_t)r * (F_FEAT / 4) + c4),
                (__attribute__((address_space(3))) v4i*)
                    (__attribute__((address_space(3))) void*)
                        (void*)(xs + r * XS_STRIDE + c4 * 4),
                0, 0);
#else
            float4 v = gx[(size_t)r * (F_FEAT / 4) + c4];
            *(float4*)(xs + r * XS_STRIDE + c4 * 4) = v;
#endif
        }
#if USE_ASYNC
        __builtin_amdgcn_s_wait_asynccnt(0);
#endif
    }
    __syncthreads();

    // ---- per-wave WMMA loop: 16 batch rows/wave, 2 scope-pairs per iteration ----
    const int lane = tid & 31;
    const int wave = tid >> 5;
    const int col  = lane & 15;          // batch column within wave tile / A row idx
    const int hi   = lane >> 4;          // 0: K0/K1 & even-p lanes, 1: K2/K3 & odd-p lanes
    const int arow = col;                // A-matrix row this lane carries
    const bool lo_active = (arow < 8);   // rows 0-7 -> pair p0, rows 8-15 -> pair p1
    const int mloc  = arow & 7;
    const int rowhi = arow >> 3;

    const float* xrow   = xs + (wave * 16 + col) * XS_STRIDE;
    const float* aBase  = coefA + (rowhi * 8 + mloc) * 4 + hi * 2; // +64 floats per q
    const float* cBase  = cTab + hi * 8;                           // +16 floats per q
    const int*   sBase  = sidx + hi;                               // sidx[4q+hi], sidx[4q+2+hi]

    float acc = 0.f;   // running sum of log2-domain logsumexp values

    for (int q = 0; q < P_PAIR / 2; ++q) {
        // B matrix: features of 16 batch rows for pair p (lanes<16: x0 side, >=16: x1 side)
        const int s0 = sBase[4 * q];
        const int s1 = sBase[4 * q + 2];
        const float xv0 = xrow[s0];
        const float xv1 = xrow[s1];
        v2f b1; b1[0] = xv0 * xv0; b1[1] = xv0;
        v2f b2; b2[0] = xv1 * xv1; b2[1] = xv1;

        // A matrix: coefficients; zero the inactive row-half per WMMA
        v2f ar = *(const v2f*)(aBase + 64 * q);
        v2f a1, a2;
        a1[0] = lo_active ? ar[0] : 0.f;  a1[1] = lo_active ? ar[1] : 0.f;
        a2[0] = lo_active ? 0.f : ar[0];  a2[1] = lo_active ? 0.f : ar[1];

        // C preloaded with per-(p,m) constants
        const float* cp = cBase + 16 * q;
        v8f cacc;
#pragma unroll
        for (int j = 0; j < 8; ++j) cacc[j] = cp[j];

        cacc = __builtin_amdgcn_wmma_f32_16x16x4_f32(false, a1, false, b1,
                                                     (short)0, cacc, false, false);
        cacc = __builtin_amdgcn_wmma_f32_16x16x4_f32(false, a2, false, b2,
                                                     (short)0, cacc, false, false);

        // thread-local logsumexp over the 8 mixture components (base-2 domain)
        float mx = fmaxf(fmaxf(fmaxf(cacc[0], cacc[1]), fmaxf(cacc[2], cacc[3])),
                         fmaxf(fmaxf(cacc[4], cacc[5]), fmaxf(cacc[6], cacc[7])));
        float ssum = 0.f;
#pragma unroll
        for (int j = 0; j < 8; ++j) ssum += EXP2F(cacc[j] - mx);
        acc += mx + LOG2F(ssum);
    }

    // lanes 0-15 accumulated even pairs, lanes 16-31 odd pairs: fold halves
    float other = __shfl_xor(acc, 16, 32);
    if (hi == 0) {
        out[rowbase + wave * 16 + col] = (acc + other) * LN2;
    }
}

extern "C" void kernel_launch(void* const* d_in, const int* in_sizes, int n_in,
                              void* d_out, int out_size, void* d_ws, size_t ws_size,
                              hipStream_t stream) {
    const float*     x      = (const float*)d_in[0];
    const float*     mean   = (const float*)d_in[1];
    const float*     stdv   = (const float*)d_in[2];
    const float*     w      = (const float*)d_in[3];
    const long long* scopes = (const long long*)d_in[4];
    float* out = (float*)d_out;

    dim3 grid(B_TOTAL / ROWS_PER_BLOCK);
    spn_wmma_kernel<<<grid, THREADS, LDS_TOTAL, stream>>>(x, mean, stdv, w, scopes, out);
}